// PairwiseCrossSimilarity_26877905339203
// MI455X (gfx1250) — compile-verified
//
#include <hip/hip_runtime.h>
#include <hip/hip_bf16.h>

typedef __attribute__((ext_vector_type(2))) float v2f;
typedef __attribute__((ext_vector_type(4))) float v4f;
typedef __attribute__((ext_vector_type(8))) float v8f;

// Problem constants (from reference)
#define BSZ   16
#define BAG   50
#define XDIM  100      // bag * cobj
#define DDIM  512
#define HDIM  512
#define NT    512      // num_top

// ---------------------------------------------------------------------------
// Stage 1: h[p][b][x][:] = fea[b][p][x][:] @ W_p   (fp32 WMMA 16x16x4)
// One wave per 16x16 output tile. 7 M-tiles cover 100 rows; out-of-range rows
// are CLAMPED to row 99 for the load (always-valid address) and the fragment
// is zeroed with a VALU select — no EXEC-mask branches in the K loop.
// jobs = 2 * 16 * 7 * 32 = 7168 waves = 896 blocks x 8 waves (exact).
// ---------------------------------------------------------------------------
__global__ void __launch_bounds__(256)
gemm_h_kernel(const float* __restrict__ fea,
              const float* __restrict__ W0,
              const float* __restrict__ W1,
              float* __restrict__ h) {
  const int lane  = threadIdx.x & 31;
  const int w     = blockIdx.x * (blockDim.x >> 5) + (threadIdx.x >> 5);
  const int ntile = w & 31;
  const int mtile = (w >> 5) % 7;
  const int b     = (w / 224) & 15;
  const int p     = w / 3584;

  const float* A = fea + (size_t)(b * 2 + p) * XDIM * DDIM;   // [100,512]
  const float* W = (p == 0) ? W0 : W1;                        // [512,512]
  float*       H = h + (size_t)(p * BSZ + b) * XDIM * HDIM;   // [100,512]

  // A fragment: lane L -> row M = L%16; VGPR pair holds K = 2*(L/16)+{0,1}
  const int mrow   = mtile * 16 + (lane & 15);
  const int kgrp   = (lane >> 4) * 2;
  const int ncol   = ntile * 16 + (lane & 15);
  const bool mvalid = (mrow < XDIM);
  const int mclamp = mvalid ? mrow : (XDIM - 1);      // always-legal address
  const float* Arow = A + (size_t)mclamp * DDIM;
  const float  msel = mvalid ? 1.0f : 0.0f;           // lane-constant

  v8f acc = {};
  for (int k = 0; k < DDIM; k += 4) {
    v2f a = *(const v2f*)(Arow + k + kgrp);           // unconditional b64 load
    a.x *= msel;                                      // v_cndmask/v_mul, no branch
    a.y *= msel;
    v2f bf;
    bf.x = W[(size_t)(k + kgrp + 0) * HDIM + ncol];
    bf.y = W[(size_t)(k + kgrp + 1) * HDIM + ncol];
    // D = A(16x4) * B(4x16) + C ; fp32 throughout
    acc = __builtin_amdgcn_wmma_f32_16x16x4_f32(false, a, false, bf,
                                                (short)0, acc, false, false);
  }

  // C/D layout: VGPR r, lanes 0-15 -> M = r, lanes 16-31 -> M = r+8
#pragma unroll
  for (int r = 0; r < 8; ++r) {
    int m = mtile * 16 + r + 8 * (lane >> 4);
    if (m < XDIM) H[(size_t)m * HDIM + ncol] = acc[r];
  }
}

// ---------------------------------------------------------------------------
// Stage 2: full[b][x][y] = sum_h v[h] * relu(h0[b,x,h] + h1[b,y,h] + bias[h])
// One wave per (b,x): h0 row / bias / v held in registers (16 elems/lane),
// inner loop over y streams h1 rows (L2-resident), wave32 shuffle reduction.
// waves = 16*100 = 1600 = 200 blocks x 8 waves (exact).
// ---------------------------------------------------------------------------
__global__ void __launch_bounds__(256)
relation_full_kernel(const float* __restrict__ h,
                     const float* __restrict__ bias,
                     const float* __restrict__ v,
                     float* __restrict__ full) {
  const int lane = threadIdx.x & 31;
  const int w = blockIdx.x * (blockDim.x >> 5) + (threadIdx.x >> 5);
  const int b = w / XDIM;
  const int x = w % XDIM;

  const float* h0  = h + ((size_t)b * XDIM + x) * HDIM;
  const float* h1b = h + (size_t)BSZ * XDIM * HDIM + (size_t)b * XDIM * HDIM;

  float h0r[16], bv[16], vv[16];
#pragma unroll
  for (int j = 0; j < 16; ++j) {
    int idx = lane + 32 * j;
    h0r[j] = h0[idx];
    bv[j]  = bias[idx];
    vv[j]  = v[idx];
  }

  float* outrow = full + ((size_t)b * XDIM + x) * XDIM;
  for (int y = 0; y < XDIM; ++y) {
    const float* h1 = h1b + (size_t)y * HDIM;
    float acc = 0.f;
#pragma unroll
    for (int j = 0; j < 16; ++j) {
      float s = h0r[j] + h1[lane + 32 * j] + bv[j];
      s = fmaxf(s, 0.f);
      acc = fmaf(s, vv[j], acc);
    }
#pragma unroll
    for (int off = 16; off > 0; off >>= 1)
      acc += __shfl_xor(acc, off, 32);
    if (lane == 0) outrow[y] = acc;
  }
}

// ---------------------------------------------------------------------------
// Stage 3: pooled[b,i,j] = sum_{c0,c1} full[b, ind0[b,i,c0]+50c0,
//                                              ind1[b,j,c1]+50c1]
// plus the (i,j) pair grid. One block (128 thr) per (b,i) row; each thread
// covers 4 consecutive j -> 128-bit NT stores. Output = 50.3 MB (HBM bound).
// blocks = 16*512 = 8192 (exact).
// ---------------------------------------------------------------------------
__global__ void __launch_bounds__(128)
gather_pool_kernel(const float* __restrict__ full,
                   const int* __restrict__ ind0,
                   const int* __restrict__ ind1,
                   float* __restrict__ out) {
  const int bi = blockIdx.x;
  const int b  = bi >> 9;
  const int i  = bi & 511;
  const int j0 = threadIdx.x * 4;

  const int r0 = ind0[((size_t)b * NT + i) * 2 + 0];
  const int r1 = ind0[((size_t)b * NT + i) * 2 + 1] + BAG;
  const float* F0 = full + ((size_t)b * XDIM + r0) * XDIM;
  const float* F1 = full + ((size_t)b * XDIM + r1) * XDIM;

  const int* idx = ind1 + ((size_t)b * NT + j0) * 2;  // 8 consecutive ints
  int4 ia = *(const int4*)(idx + 0);
  int4 ib = *(const int4*)(idx + 4);

  int c0[4] = {ia.x, ia.z, ib.x, ib.z};
  int c1[4] = {ia.y + BAG, ia.w + BAG, ib.y + BAG, ib.w + BAG};

  float vals[4];
#pragma unroll
  for (int q = 0; q < 4; ++q)
    vals[q] = F0[c0[q]] + F0[c1[q]] + F1[c0[q]] + F1[c1[q]];

  v4f sc = {vals[0], vals[1], vals[2], vals[3]};
  size_t obase = ((size_t)b * NT + i) * NT + j0;
  __builtin_nontemporal_store(sc, (v4f*)(out + obase));

  // pairs output: concatenated after scores, as floats (exact for 0..511)
  float* pbase = out + (size_t)BSZ * NT * NT;
  size_t pidx = (((size_t)b * NT + i) * NT + j0) * 2;
  float fi = (float)i;
  v4f p0 = {fi, (float)(j0 + 0), fi, (float)(j0 + 1)};
  v4f p1 = {fi, (float)(j0 + 2), fi, (float)(j0 + 3)};
  __builtin_nontemporal_store(p0, (v4f*)(pbase + pidx));
  __builtin_nontemporal_store(p1, (v4f*)(pbase + pidx + 4));
}

// ---------------------------------------------------------------------------
extern "C" void kernel_launch(void* const* d_in, const int* in_sizes, int n_in,
                              void* d_out, int out_size, void* d_ws, size_t ws_size,
                              hipStream_t stream) {
  const float* fea  = (const float*)d_in[0];
  const int*   ind0 = (const int*)d_in[1];
  const int*   ind1 = (const int*)d_in[2];
  const float* W0   = (const float*)d_in[3];
  const float* W1   = (const float*)d_in[4];
  const float* bias = (const float*)d_in[5];
  const float* v    = (const float*)d_in[6];

  float* ws   = (float*)d_ws;
  float* h    = ws;                                   // [2][16][100][512] = 6.55 MB
  float* full = ws + (size_t)2 * BSZ * XDIM * HDIM;   // [16][100][100]    = 640 KB

  gemm_h_kernel<<<896, 256, 0, stream>>>(fea, W0, W1, h);
  relation_full_kernel<<<200, 256, 0, stream>>>(h, bias, v, full);
  gather_pool_kernel<<<BSZ * NT, 128, 0, stream>>>(full, ind0, ind1, (float*)d_out);
}